// HydraLoRA_67577015435775
// MI455X (gfx1250) — compile-verified
//
#include <hip/hip_runtime.h>
#include <math.h>

typedef __attribute__((ext_vector_type(2))) float v2f;
typedef __attribute__((ext_vector_type(8))) float v8f;
typedef __attribute__((vector_size(4 * sizeof(int)))) int vi4;
typedef __attribute__((vector_size(2 * sizeof(int)))) int vi2;

#define T_TOKENS 8192
#define DIM_IN   4096
#define DIM_OUT  4096
#define NEXP     8
#define RANK     32
#define ER       256
#define SCALING  1.0f          // alpha/r = 32/32
#define KC       64            // K chunk
#define XS       68            // row stride (dwords) for 64-wide tiles: %64==4 -> conflict-free, rows 16B aligned
#define BS       272           // row stride (dwords) for 256-wide tiles: %64==16 -> conflict-free, rows 16B aligned

// ---- CDNA5 async global->LDS copies (ASYNCcnt), with synchronous fallback ----
#if defined(__has_builtin)
#  if __has_builtin(__builtin_amdgcn_global_load_async_to_lds_b128) && \
      __has_builtin(__builtin_amdgcn_global_load_async_to_lds_b64)  && \
      __has_builtin(__builtin_amdgcn_s_wait_asynccnt)
#    define USE_ASYNC 1
#  endif
#endif

__device__ __forceinline__ void async_copy_b128(const float* g, float* l) {
#ifdef USE_ASYNC
  __builtin_amdgcn_global_load_async_to_lds_b128(
      (__attribute__((address_space(1))) vi4*)g,
      (__attribute__((address_space(3))) vi4*)l, 0, 0);
#else
  *(float4*)l = *(const float4*)g;
#endif
}
__device__ __forceinline__ void async_copy_b64(const float* g, float* l) {
#ifdef USE_ASYNC
  __builtin_amdgcn_global_load_async_to_lds_b64(
      (__attribute__((address_space(1))) vi2*)g,
      (__attribute__((address_space(3))) vi2*)l, 0, 0);
#else
  *(float2*)l = *(const float2*)g;
#endif
}
#ifdef USE_ASYNC
#  define WAIT_ASYNC(n) __builtin_amdgcn_s_wait_asynccnt(n)
#else
#  define WAIT_ASYNC(n) ((void)0)
#endif

#define WMMA_F32(a, b, c) \
  __builtin_amdgcn_wmma_f32_16x16x4_f32(false, (a), false, (b), (short)0, (c), false, false)

// ---------------------------------------------------------------------------
// Kernel 1: transpose Bw [E, OUT, R] -> BwT [E*R, OUT]
// ---------------------------------------------------------------------------
__global__ __launch_bounds__(256) void bw_transpose_kernel(
    const float* __restrict__ Bw, float* __restrict__ BwT)
{
  __shared__ float tile[32][33];
  const int e  = blockIdx.y;
  const int o0 = blockIdx.x * 32;
  const int t  = threadIdx.x;
  const int r  = t & 31;
  const int oi = t >> 5;
#pragma unroll
  for (int j = 0; j < 4; ++j)
    tile[oi + 8 * j][r] =
        Bw[(size_t)e * DIM_OUT * RANK + (size_t)(o0 + oi + 8 * j) * RANK + r];
  __syncthreads();
  const int ol = t & 31;
  const int ri = t >> 5;
#pragma unroll
  for (int j = 0; j < 4; ++j)
    BwT[(size_t)(e * RANK + ri + 8 * j) * DIM_OUT + o0 + ol] = tile[ol][ri + 8 * j];
}

// ---------------------------------------------------------------------------
// Kernel 2: fused router + all-expert LoRA-A projection, M=32 tokens/WG.
// Double-buffered async LDS pipeline over K. Writes gh[t][er] =
// SCALING * gate(t, er/32) * h  (0 for unselected experts).
// Per-buffer floats: xs[32][XS]=2176 | as[256][XS]=17408 | wsh[8][XS]=544 => 20128
// ---------------------------------------------------------------------------
#define K2_BUF 20128
__global__ __launch_bounds__(256) void lora_router_h_kernel(
    const float* __restrict__ x, const float* __restrict__ A,
    const float* __restrict__ Wr, const float* __restrict__ br,
    float* __restrict__ gh)
{
  extern __shared__ float smem[];
  const int t  = threadIdx.x;
  const int wv = t >> 5;              // wave == expert owned by this wave
  const int l  = t & 31;
  const int m  = l & 15;
  const int hi = l >> 4;
  const int t0 = blockIdx.x * 32;
  const int rm = t & 31;              // router token
  const int re = t >> 5;              // router expert

  float* lm  = smem + 2 * K2_BUF;     // [32][8] logits
  float* wgt = lm + 256;              // [32][8] gate*SCALING

  v8f c00 = {}, c01 = {}, c10 = {}, c11 = {};
  float racc = 0.f;

  auto issue = [&](float* buf, int k0) {          // 19 async ops / wave
    float* xs  = buf;
    float* as  = buf + 2176;
    float* wsh = buf + 19584;
#pragma unroll
    for (int j = 0; j < 2; ++j) {
      const int flat = t + 256 * j, row = flat >> 4, c4 = flat & 15;
      async_copy_b128(x + (size_t)(t0 + row) * DIM_IN + k0 + c4 * 4,
                      xs + row * XS + c4 * 4);
    }
#pragma unroll
    for (int j = 0; j < 16; ++j) {
      const int flat = t + 256 * j, er = flat >> 4, c4 = flat & 15;
      async_copy_b128(A + (size_t)er * DIM_IN + k0 + c4 * 4,
                      as + er * XS + c4 * 4);
    }
    async_copy_b64(Wr + (size_t)re * DIM_IN + k0 + rm * 2,
                   wsh + re * XS + rm * 2);
  };

  auto compute = [&](const float* buf) {
    const float* xs  = buf;
    const float* as  = buf + 2176;
    const float* wsh = buf + 19584;
    const int er0 = wv * 32 + m, er1 = er0 + 16;
#pragma unroll 8
    for (int s = 0; s < 16; ++s) {
      const int ka = s * 4 + 2 * hi;
      v2f a0; a0.x = xs[m * XS + ka];          a0.y = xs[m * XS + ka + 1];
      v2f a1; a1.x = xs[(16 + m) * XS + ka];   a1.y = xs[(16 + m) * XS + ka + 1];
      v2f b0; b0.x = as[er0 * XS + ka];        b0.y = as[er0 * XS + ka + 1];
      v2f b1; b1.x = as[er1 * XS + ka];        b1.y = as[er1 * XS + ka + 1];
      c00 = WMMA_F32(a0, b0, c00);  c01 = WMMA_F32(a0, b1, c01);
      c10 = WMMA_F32(a1, b0, c10);  c11 = WMMA_F32(a1, b1, c11);
    }
#pragma unroll 16
    for (int kk = 0; kk < KC; ++kk)
      racc = fmaf(xs[rm * XS + kk], wsh[re * XS + kk], racc);
  };

  issue(smem, 0);
  for (int it = 0; it < DIM_IN / KC - 1; ++it) {
    issue(smem + ((it + 1) & 1) * K2_BUF, (it + 1) * KC);
    WAIT_ASYNC(19);                  // previous chunk complete, next in flight
    __syncthreads();
    compute(smem + (it & 1) * K2_BUF);
    __syncthreads();
  }
  WAIT_ASYNC(0);
  __syncthreads();
  compute(smem + ((DIM_IN / KC - 1) & 1) * K2_BUF);

  // ---- router finalize: max-sub, clamp, top-2, softmax ----
  lm[rm * 8 + re] = racc + br[re];
  __syncthreads();
  if (t < 32) {
    float v[NEXP];
    float mx = -3.0e38f;
#pragma unroll
    for (int e = 0; e < NEXP; ++e) { v[e] = lm[t * 8 + e]; mx = fmaxf(mx, v[e]); }
    int i1 = 0; float bv1 = -3.0e38f;
#pragma unroll
    for (int e = 0; e < NEXP; ++e) {
      const float z = fminf(fmaxf(v[e] - mx, -20.f), 20.f);
      v[e] = z;
      if (z > bv1) { bv1 = z; i1 = e; }
    }
    int i2 = 0; float bv2 = -3.0e38f;
#pragma unroll
    for (int e = 0; e < NEXP; ++e)
      if (e != i1 && v[e] > bv2) { bv2 = v[e]; i2 = e; }
    const float e1 = expf(bv1), e2 = expf(bv2);
    const float inv = 1.f / (e1 + e2);
#pragma unroll
    for (int e = 0; e < NEXP; ++e) wgt[t * 8 + e] = 0.f;
    wgt[t * 8 + i1] = SCALING * e1 * inv;
    wgt[t * 8 + i2] = SCALING * e2 * inv;
  }
  __syncthreads();

  const int er0 = wv * 32 + m;
#pragma unroll
  for (int j = 0; j < 8; ++j) {
    {
      const int mm = j + 8 * hi;
      const float f = wgt[mm * 8 + wv];
      gh[(size_t)(t0 + mm) * ER + er0]      = c00[j] * f;
      gh[(size_t)(t0 + mm) * ER + er0 + 16] = c01[j] * f;
    }
    {
      const int mm = 16 + j + 8 * hi;
      const float f = wgt[mm * 8 + wv];
      gh[(size_t)(t0 + mm) * ER + er0]      = c10[j] * f;
      gh[(size_t)(t0 + mm) * ER + er0 + 16] = c11[j] * f;
    }
  }
}

// ---------------------------------------------------------------------------
// Kernel 3: out[T][4096] = gh[T][256] @ BwT[256][4096], M=64 x N=256 per WG.
// Per-buffer floats: hs[64][XS]=4352 | bs[64][BS]=17408 => 21760
// ---------------------------------------------------------------------------
#define K3_BUF 21760
__global__ __launch_bounds__(256) void lora_out_kernel(
    const float* __restrict__ gh, const float* __restrict__ BwT,
    float* __restrict__ out)
{
  extern __shared__ float smem[];
  const int t  = threadIdx.x;
  const int wv = t >> 5;
  const int l  = t & 31;
  const int m  = l & 15;
  const int hi = l >> 4;
  const int t0    = (blockIdx.x >> 4) * 64;
  const int nbase = (blockIdx.x & 15) * 256;

  v8f c[4][2] = {};

  auto issue = [&](float* buf, int k0) {          // 20 async ops / wave
    float* hs = buf;
    float* bs = buf + 4352;
#pragma unroll
    for (int j = 0; j < 4; ++j) {
      const int flat = t + 256 * j, row = flat >> 4, c4 = flat & 15;
      async_copy_b128(gh + (size_t)(t0 + row) * ER + k0 + c4 * 4,
                      hs + row * XS + c4 * 4);
    }
#pragma unroll
    for (int j = 0; j < 16; ++j) {
      const int flat = t + 256 * j, kk = flat >> 6, c4 = flat & 63;
      async_copy_b128(BwT + (size_t)(k0 + kk) * DIM_OUT + nbase + c4 * 4,
                      bs + kk * BS + c4 * 4);
    }
  };

  auto compute = [&](const float* buf) {
    const float* hs = buf;
    const float* bs = buf + 4352;
    const int n0 = wv * 32 + m, n1 = n0 + 16;
#pragma unroll 4
    for (int s = 0; s < 16; ++s) {
      const int ka = s * 4 + 2 * hi;
      v2f b0; b0.x = bs[ka * BS + n0]; b0.y = bs[(ka + 1) * BS + n0];
      v2f b1; b1.x = bs[ka * BS + n1]; b1.y = bs[(ka + 1) * BS + n1];
#pragma unroll
      for (int mi = 0; mi < 4; ++mi) {
        v2f a; a.x = hs[(mi * 16 + m) * XS + ka]; a.y = hs[(mi * 16 + m) * XS + ka + 1];
        c[mi][0] = WMMA_F32(a, b0, c[mi][0]);
        c[mi][1] = WMMA_F32(a, b1, c[mi][1]);
      }
    }
  };

  issue(smem, 0);
  for (int it = 0; it < ER / KC - 1; ++it) {
    issue(smem + ((it + 1) & 1) * K3_BUF, (it + 1) * KC);
    WAIT_ASYNC(20);
    __syncthreads();
    compute(smem + (it & 1) * K3_BUF);
    __syncthreads();
  }
  WAIT_ASYNC(0);
  __syncthreads();
  compute(smem + ((ER / KC - 1) & 1) * K3_BUF);

  const int n0 = wv * 32 + m;
#pragma unroll
  for (int mi = 0; mi < 4; ++mi)
#pragma unroll
    for (int j = 0; j < 8; ++j) {
      const int mm = mi * 16 + j + 8 * hi;
      out[(size_t)(t0 + mm) * DIM_OUT + nbase + n0]      = c[mi][0][j];
      out[(size_t)(t0 + mm) * DIM_OUT + nbase + n0 + 16] = c[mi][1][j];
    }
}

// ---------------------------------------------------------------------------
extern "C" void kernel_launch(void* const* d_in, const int* in_sizes, int n_in,
                              void* d_out, int out_size, void* d_ws, size_t ws_size,
                              hipStream_t stream) {
  (void)in_sizes; (void)n_in; (void)out_size; (void)ws_size;
  const float* x  = (const float*)d_in[0];   // [4,2048,4096]
  const float* A  = (const float*)d_in[1];   // [8,32,4096]
  const float* Bw = (const float*)d_in[2];   // [8,4096,32]
  const float* Wr = (const float*)d_in[3];   // [8,4096]
  const float* br = (const float*)d_in[4];   // [8]
  float* out = (float*)d_out;                // [4,2048,4096]

  float* gh  = (float*)d_ws;                                                  // [8192][256]
  float* BwT = (float*)((char*)d_ws + (size_t)T_TOKENS * ER * sizeof(float)); // [256][4096]

  const size_t lds_2 = (size_t)(2 * K2_BUF + 512) * sizeof(float);  // 163072 B
  const size_t lds_3 = (size_t)(2 * K3_BUF) * sizeof(float);        // 174080 B

  bw_transpose_kernel<<<dim3(DIM_OUT / 32, NEXP), 256, 0, stream>>>(Bw, BwT);
  lora_router_h_kernel<<<T_TOKENS / 32, 256, lds_2, stream>>>(x, A, Wr, br, gh);
  lora_out_kernel<<<(T_TOKENS / 64) * (DIM_OUT / 256), 256, lds_3, stream>>>(gh, BwT, out);
}